// GPUErrorModel_50611894616858
// MI455X (gfx1250) — compile-verified
//
#include <hip/hip_runtime.h>
#include <stdint.h>

typedef unsigned int u32;
typedef unsigned long long u64;

// ---------------------------------------------------------------------------
// Event record (one per error event) and reduction metadata, kept in d_ws.
// ---------------------------------------------------------------------------
struct Event {
    int type;    // 0 single_bit, 1 byte_aligned, 2 non_byte, 3 whole_chip, 4 pin
    u32 a;       // idx / start / whole_chip key0 / pin mask
    u32 b;       // xor mask / whole_chip key1
    u32 m[8];    // non_byte: per-byte masks
};

struct Meta {
    int wc_active;   // 1 if a whole_chip event occurred
    int lastL;       // index of last whole_chip event, -1 if none
    u32 wck0, wck1;  // key of that whole_chip event
    u32 pinmask;     // XOR of all pin masks after lastL
};

// ---------------------------------------------------------------------------
// Threefry-2x32-20 (exact JAX algorithm: 5 groups of 4 rounds)
// ---------------------------------------------------------------------------
__device__ __forceinline__ u32 rotl32(u32 v, int s) { return (v << s) | (v >> (32 - s)); }

__device__ __forceinline__ void tf2x32(u32 k0, u32 k1, u32 x0, u32 x1, u32& o0, u32& o1) {
    const u32 ks2 = 0x1BD11BDAu ^ k0 ^ k1;
    x0 += k0; x1 += k1;
#define TF_ROUND(r) { x0 += x1; x1 = rotl32(x1, (r)); x1 ^= x0; }
    TF_ROUND(13) TF_ROUND(15) TF_ROUND(26) TF_ROUND(6)
    x0 += k1;  x1 += ks2 + 1u;
    TF_ROUND(17) TF_ROUND(29) TF_ROUND(16) TF_ROUND(24)
    x0 += ks2; x1 += k0 + 2u;
    TF_ROUND(13) TF_ROUND(15) TF_ROUND(26) TF_ROUND(6)
    x0 += k0;  x1 += k1 + 3u;
    TF_ROUND(17) TF_ROUND(29) TF_ROUND(16) TF_ROUND(24)
    x0 += k1;  x1 += ks2 + 4u;
    TF_ROUND(13) TF_ROUND(15) TF_ROUND(26) TF_ROUND(6)
    x0 += ks2; x1 += k0 + 5u;
#undef TF_ROUND
    o0 = x0; o1 = x1;
}

// Element j of random_bits(key,32,(2N,)) as used by jax.random.split(key, N):
// counters are iota(2N) split into halves x0=[0..N), x1=[N..2N).
__device__ u32 split_flat(u32 k0, u32 k1, int N, int j) {
    u32 y0, y1;
    if (j < N) { tf2x32(k0, k1, (u32)j, (u32)(j + N), y0, y1); return y0; }
    tf2x32(k0, k1, (u32)(j - N), (u32)j, y0, y1); return y1;
}

// split(key, 2): key0=(y0(0,2), y0(1,3)), key1=(y1(0,2), y1(1,3))
__device__ void split2(u32 k0, u32 k1, u32& a0, u32& a1, u32& b0, u32& b1) {
    u32 p0, p1, q0, q1;
    tf2x32(k0, k1, 0u, 2u, p0, p1);
    tf2x32(k0, k1, 1u, 3u, q0, q1);
    a0 = p0; a1 = q0; b0 = p1; b1 = q1;
}

// split(key, 3): flat=[y0(0,3),y0(1,4),y0(2,5),y1(0,3),y1(1,4),y1(2,5)]
__device__ void split3(u32 k0, u32 k1, u32 o[6]) {
    u32 y0, y1;
    tf2x32(k0, k1, 0u, 3u, y0, y1); o[0] = y0; o[3] = y1;
    tf2x32(k0, k1, 1u, 4u, y0, y1); o[1] = y0; o[4] = y1;
    tf2x32(k0, k1, 2u, 5u, y0, y1); o[2] = y0; o[5] = y1;
}

// jax.random.randint(key, (), 0, span): draws bits shape (2,) -> one threefry
// of counter (0,1); value = ((hi%span)*(2^32%span) + lo%span) % span
__device__ u32 randint_scalar(u32 k0, u32 k1, u32 span) {
    u32 hi, lo;
    tf2x32(k0, k1, 0u, 1u, hi, lo);
    u32 mult = (u32)(4294967296ull % (u64)span);
    u64 off = ((u64)(hi % span) * (u64)mult + (u64)(lo % span)) % (u64)span;
    return (u32)off;
}

// random_bits(key, 32, ()) : odd count padded -> counter (0,0), take y0
__device__ u32 bits_scalar(u32 k0, u32 k1) {
    u32 a, b; tf2x32(k0, k1, 0u, 0u, a, b); return a;
}

// ---------------------------------------------------------------------------
// Kernel 1: compute the 268 event descriptors (replicates the JAX scan keys)
// ---------------------------------------------------------------------------
__global__ void ev_kernel(Event* __restrict__ ev, int EB, int n) {
    int i = blockIdx.x * blockDim.x + threadIdx.x;
    if (i >= EB) return;

    // keys = split(key(42), EB); key(42) data = (0, 42)
    const u32 K0 = 0u, K1 = 42u;
    u32 e0 = split_flat(K0, K1, EB, 2 * i);
    u32 e1 = split_flat(K0, K1, EB, 2 * i + 1);

    // k_type, k_apply = split(keys[i])
    u32 t0, t1, a0, a1;
    split2(e0, e1, t0, t1, a0, a1);

    // t = choice(k_type, 5, p=WEIGHTS): r = cuml[-1]*(1-uniform); searchsorted left
    u32 ub = bits_scalar(t0, t1);
    float u = __uint_as_float(0x3f800000u | (ub >> 9)) - 1.0f;
    float c0 = 0.7398f, c1 = c0 + 0.2256f, c2 = c1 + 0.009f,
          c3 = c2 + 0.0223f, c4 = c3 + 0.0019f;
    float r = c4 * (1.0f - u);
    int t = (int)(c0 < r) + (int)(c1 < r) + (int)(c2 < r) + (int)(c3 < r) + (int)(c4 < r);

    Event E;
    E.type = t; E.a = 0u; E.b = 0u;
    for (int j = 0; j < 8; ++j) E.m[j] = 0u;

    if (t == 0) {                       // single_bit
        u32 s0, s1, s2, s3;
        split2(a0, a1, s0, s1, s2, s3);
        E.a = randint_scalar(s0, s1, (u32)n);
        E.b = 1u << randint_scalar(s2, s3, 8u);
    } else if (t == 1) {                // byte_aligned
        u32 o[6]; split3(a0, a1, o);
        u32 idx = randint_scalar(o[0], o[1], (u32)n);
        u32 nb  = 2u + randint_scalar(o[2], o[3], 7u);
        // permutation(k3, 8): one shuffle round: key',sub = split(k3); sort arange by bits(sub,(8,))
        u32 d0, d1, s0, s1;
        split2(o[4], o[5], d0, d1, s0, s1);
        u32 sk[8]; int perm[8];
        for (int j = 0; j < 4; ++j) {
            u32 y0, y1; tf2x32(s0, s1, (u32)j, (u32)(j + 4), y0, y1);
            sk[j] = y0; sk[j + 4] = y1;
        }
        for (int j = 0; j < 8; ++j) perm[j] = j;
        for (int j = 1; j < 8; ++j) {   // stable insertion sort (sort_key_val)
            u32 kv = sk[j]; int pv = perm[j]; int q = j - 1;
            while (q >= 0 && sk[q] > kv) { sk[q + 1] = sk[q]; perm[q + 1] = perm[q]; --q; }
            sk[q + 1] = kv; perm[q + 1] = pv;
        }
        u32 mask = 0u;
        for (u32 j = 0; j < nb; ++j) mask |= (1u << perm[j]);
        E.a = idx; E.b = mask;
    } else if (t == 2) {                // non_byte: parity of 64 random bit hits
        u32 o[6]; split3(a0, a1, o);
        u32 start = randint_scalar(o[0], o[1], (u32)(n - 7));
        u32 nb = 2u + randint_scalar(o[2], o[3], 63u);
        u32 par0 = 0u, par1 = 0u;
        for (u32 j = 0; j < nb; ++j) {  // bit_idxs[j] = y1 of tf(k3,(j,j+64)) & 63
            u32 y0, y1; tf2x32(o[4], o[5], j, j + 64u, y0, y1);
            u32 bi = y1 & 63u;
            if (bi < 32u) par0 ^= (1u << bi); else par1 ^= (1u << (bi - 32u));
        }
        E.a = start;
        for (int bt = 0; bt < 8; ++bt) {
            u32 mask = 0u;
            for (int j = 0; j < 8; ++j) {
                int p = bt * 8 + j;
                u32 pv = ((p < 32) ? (par0 >> p) : (par1 >> (p - 32))) & 1u;
                mask |= pv << j;
            }
            E.m[bt] = mask;
        }
    } else if (t == 3) {                // whole_chip: remember k_apply
        E.a = a0; E.b = a1;
    } else {                            // pin
        E.a = 1u << randint_scalar(a0, a1, 8u);
    }
    ev[i] = E;
}

// ---------------------------------------------------------------------------
// Kernel 2: sequential reduction over 268 events (trivial)
// ---------------------------------------------------------------------------
__global__ void reduce_kernel(const Event* __restrict__ ev, Meta* __restrict__ meta, int EB) {
    if (blockIdx.x != 0 || threadIdx.x != 0) return;
    int L = -1;
    for (int i = 0; i < EB; ++i) if (ev[i].type == 3) L = i;
    u32 pin = 0u;
    for (int i = L + 1; i < EB; ++i) if (ev[i].type == 4) pin ^= ev[i].a;
    meta->wc_active = (L >= 0) ? 1 : 0;
    meta->lastL = L;
    meta->wck0 = (L >= 0) ? ev[L].a : 0u;
    meta->wck1 = (L >= 0) ? ev[L].b : 0u;
    meta->pinmask = pin;
}

// ---------------------------------------------------------------------------
// Kernel 3: bulk base generation / streaming copy+xor (the roofline phase).
// Copy path uses the gfx1250 async global->LDS engine (ASYNCcnt) with a
// 2-stage LDS pipeline: global_load_async_to_lds_b128 + s_wait_asynccnt,
// and reads the staged data back through true ds_load (addrspace(3)).
// ---------------------------------------------------------------------------
typedef __attribute__((address_space(3))) int lds_int_t;

#define ASYNC_LOAD_B128(ldsoff, gptr)                                        \
    asm volatile("global_load_async_to_lds_b128 %0, %1, off"                 \
                 :: "v"(ldsoff), "v"(gptr) : "memory")

__global__ __launch_bounds__(256) void base_kernel(const int* __restrict__ x,
                                                   int* __restrict__ out,
                                                   const Meta* __restrict__ meta,
                                                   int n) {
    __shared__ __align__(16) int lbuf[2][256][4];   // 2 stages x 256 lanes x 16B = 8KB

    const int wc  = meta->wc_active;
    const u32 pin = meta->pinmask;
    const int n4  = n >> 2;
    const int tid    = blockIdx.x * blockDim.x + threadIdx.x;
    const int stride = gridDim.x * blockDim.x;

    if (wc) {
        // out[i] = (y1 of threefry(wc_key, (i, n+i)) & 255) ^ pin    (randint span 256)
        const u32 k0 = meta->wck0, k1 = meta->wck1;
        for (int c = tid; c < n4; c += stride) {
            int i = c * 4;
            u32 y0, y1;
            int4 v;
            tf2x32(k0, k1, (u32)(i + 0), (u32)(i + 0 + n), y0, y1); v.x = (int)((y1 & 255u) ^ pin);
            tf2x32(k0, k1, (u32)(i + 1), (u32)(i + 1 + n), y0, y1); v.y = (int)((y1 & 255u) ^ pin);
            tf2x32(k0, k1, (u32)(i + 2), (u32)(i + 2 + n), y0, y1); v.z = (int)((y1 & 255u) ^ pin);
            tf2x32(k0, k1, (u32)(i + 3), (u32)(i + 3 + n), y0, y1); v.w = (int)((y1 & 255u) ^ pin);
            ((int4*)out)[c] = v;
        }
    } else {
        // Streaming out[i] = x[i] ^ pin via async LDS staging, double buffered.
        const int4* x4 = (const int4*)x;
        lds_int_t* lp0 = (lds_int_t*)(&lbuf[0][threadIdx.x][0]);
        lds_int_t* lp1 = (lds_int_t*)(&lbuf[1][threadIdx.x][0]);
        u32 lds0 = (u32)(size_t)lp0;
        u32 lds1 = (u32)(size_t)lp1;

        int c = tid;
        if (c < n4) ASYNC_LOAD_B128(lds0, (const void*)(x4 + c));
        int stage = 0;
        for (; c < n4; c += stride) {
            int cn = c + stride;
            if (cn < n4) {
                ASYNC_LOAD_B128(stage ? lds0 : lds1, (const void*)(x4 + cn));
                asm volatile("s_wait_asynccnt 1" ::: "memory");  // current stage landed
            } else {
                asm volatile("s_wait_asynccnt 0" ::: "memory");
            }
            // ds_load path: volatile addrspace(3) reads cannot be folded/hoisted
            volatile lds_int_t* lp = stage ? lp1 : lp0;
            int4 v;
            v.x = lp[0] ^ (int)pin;
            v.y = lp[1] ^ (int)pin;
            v.z = lp[2] ^ (int)pin;
            v.w = lp[3] ^ (int)pin;
            ((int4*)out)[c] = v;
            stage ^= 1;
        }
    }
}

// ---------------------------------------------------------------------------
// Kernel 4: apply sparse XOR events after the last whole_chip (order-free)
// ---------------------------------------------------------------------------
__global__ void apply_kernel(const Event* __restrict__ ev, int* __restrict__ out,
                             const Meta* __restrict__ meta, int EB) {
    int i = blockIdx.x * blockDim.x + threadIdx.x;
    if (i >= EB) return;
    if (i <= meta->lastL) return;
    Event E = ev[i];
    if (E.type == 0 || E.type == 1) {
        atomicXor(&out[E.a], (int)E.b);
    } else if (E.type == 2) {
        for (int j = 0; j < 8; ++j) atomicXor(&out[E.a + (u32)j], (int)E.m[j]);
    }
}

// ---------------------------------------------------------------------------
extern "C" void kernel_launch(void* const* d_in, const int* in_sizes, int n_in,
                              void* d_out, int out_size, void* d_ws, size_t ws_size,
                              hipStream_t stream) {
    const int* x = (const int*)d_in[0];
    int* out = (int*)d_out;
    const int n = in_sizes[0];
    const int EB = (int)((double)n * 8.0 * 1e-6);   // 268 for n = 2^25

    Event* ev = (Event*)d_ws;
    size_t ev_bytes = ((sizeof(Event) * (size_t)EB + 255) / 256) * 256;
    Meta* meta = (Meta*)((char*)d_ws + ev_bytes);

    ev_kernel<<<(EB + 63) / 64, 64, 0, stream>>>(ev, EB, n);
    reduce_kernel<<<1, 32, 0, stream>>>(ev, meta, EB);
    base_kernel<<<2048, 256, 0, stream>>>(x, out, meta, n);
    apply_kernel<<<(EB + 63) / 64, 64, 0, stream>>>(ev, out, meta, EB);
}